// HyperGraphFusion_70514773066071
// MI455X (gfx1250) — compile-verified
//
#include <hip/hip_runtime.h>
#include <math.h>

typedef __attribute__((ext_vector_type(2))) float v2f;
typedef __attribute__((ext_vector_type(4))) float v4f;
typedef __attribute__((ext_vector_type(8))) float v8f;

#define BATCH 8
#define SEQ_LT 4096
#define SEQ_LV 4096
#define DIM 1024
#define TOPK 8

// ---------------------------------------------------------------------------
// K1: zero-fill padded outputs. 268 MB > 192 MB L2 and never read back, so
// use non-temporal 128-bit stores to avoid churning L2.
// ---------------------------------------------------------------------------
__global__ void zero_fill_kernel(v4f* __restrict__ out, long long n4) {
  long long i = (long long)blockIdx.x * blockDim.x + threadIdx.x;
  long long stride = (long long)gridDim.x * blockDim.x;
  v4f z = {0.f, 0.f, 0.f, 0.f};
  for (; i < n4; i += stride) __builtin_nontemporal_store(z, &out[i]);
}

// ---------------------------------------------------------------------------
// K2: squared L2 norm per text row (sqrt monotonic -> skip for top-k).
// One wave32 per row, non-temporal b128 loads (single-use data), shuffle
// reduction across the wave.
// ---------------------------------------------------------------------------
__global__ void row_norm2_kernel(const float* __restrict__ text,
                                 float* __restrict__ scores) {
  int wave = threadIdx.x >> 5;
  int lane = threadIdx.x & 31;
  long long row = (long long)blockIdx.x * 8 + wave;  // B*LT rows total
  const v4f* r = (const v4f*)(text + row * DIM);
  float s = 0.f;
#pragma unroll
  for (int i = 0; i < 8; ++i) {
    v4f v = __builtin_nontemporal_load(&r[i * 32 + lane]);
    s += v.x * v.x + v.y * v.y + v.z * v.z + v.w * v.w;
  }
#pragma unroll
  for (int off = 16; off > 0; off >>= 1) s += __shfl_xor(s, off, 32);
  if (lane == 0) scores[row] = s;
}

// ---------------------------------------------------------------------------
// K3: top-8 per batch by iterative masked argmax (descending, ties -> low idx,
// matching jax.lax.top_k ordering).
// ---------------------------------------------------------------------------
__global__ void topk_kernel(const float* __restrict__ scores,
                            int* __restrict__ idx_out) {
  __shared__ float vals[SEQ_LT];
  __shared__ float rmax[256];
  __shared__ int rarg[256];
  int b = blockIdx.x;
  int tid = threadIdx.x;
  const float* s = scores + (long long)b * SEQ_LT;
  for (int i = tid; i < SEQ_LT; i += 256) vals[i] = s[i];
  __syncthreads();
  for (int k = 0; k < TOPK; ++k) {
    float m = -INFINITY;
    int a = 0;
    for (int i = tid; i < SEQ_LT; i += 256) {
      float v = vals[i];
      if (v > m) { m = v; a = i; }
    }
    rmax[tid] = m;
    rarg[tid] = a;
    __syncthreads();
    for (int off = 128; off > 0; off >>= 1) {
      if (tid < off) {
        float v = rmax[tid + off];
        if (v > rmax[tid] || (v == rmax[tid] && rarg[tid + off] < rarg[tid])) {
          rmax[tid] = v;
          rarg[tid] = rarg[tid + off];
        }
      }
      __syncthreads();
    }
    if (tid == 0) {
      idx_out[b * TOPK + k] = rarg[0];
      vals[rarg[0]] = -INFINITY;
    }
    __syncthreads();
  }
}

// ---------------------------------------------------------------------------
// K4: per-batch fusion. tq = gather(text)@W^T + b via V_WMMA_F32_16X16X4_F32,
// sim/softmax in LDS, then the two rank-8 output mixes.
// Block = 512 threads = 16 waves; 64 d-tiles of 16 -> 4 tiles per wave.
// A-matrix zero padding lives in LDS (row 8 is all zeros) so the K-loop has
// no lane-dependent control flow: ds_load_b64 + global_load_b64 + v_wmma.
// ---------------------------------------------------------------------------
__global__ void __launch_bounds__(512) fusion_kernel(
    const float* __restrict__ text, const float* __restrict__ visual,
    const float* __restrict__ Wk, const float* __restrict__ bk,
    const int* __restrict__ idx, float* __restrict__ out_text,
    float* __restrict__ out_vis) {
  __shared__ float ltk[(TOPK + 1) * DIM];  // rows 0..7 = keys, row 8 = zeros
  __shared__ float lsim[TOPK * TOPK];      // 8x8 logits
  __shared__ float lhe[TOPK * TOPK];       // 8x8 softmax weights

  int b = blockIdx.x;
  int tid = threadIdx.x;
  const float* vbase = visual + (long long)b * SEQ_LV * DIM;

  // Stage gathered text keys into LDS; row 8 is the zero pad row.
  for (int j = tid; j < (TOPK + 1) * DIM; j += 512) {
    int t = j >> 10, d = j & (DIM - 1);
    ltk[j] = (t < TOPK)
                 ? text[((long long)b * SEQ_LT + idx[b * TOPK + t]) * DIM + d]
                 : 0.f;
  }
  if (tid < TOPK * TOPK) lsim[tid] = 0.f;
  __syncthreads();

  int wave = tid >> 5, lane = tid & 31;
  int m = lane & 15;            // A row / B column index for this lane
  int koff = (lane >> 4) << 1;  // lanes 16-31 carry K+2,K+3
  // A rows 8..15 read the shared zero row -> address select, not data select.
  const float* arow = ltk + ((m < TOPK) ? m : TOPK) * DIM;

  for (int ti = 0; ti < 4; ++ti) {
    int dt = wave * 4 + ti;  // d-tile 0..63
    const float* Wr = Wk + (long long)(dt * 16 + m) * DIM;  // B[k][n]=W[n][k]
    v8f acc = {};
    for (int k0 = 0; k0 < DIM; k0 += 4) {
      int kk = k0 + koff;  // even -> 8-byte aligned
      v2f av = *(const v2f*)(arow + kk);  // ds_load_b64
      v2f bv = *(const v2f*)(Wr + kk);    // global_load_b64
      acc = __builtin_amdgcn_wmma_f32_16x16x4_f32(
          false, av, false, bv, (short)0, acc, false, false);
    }
    // C layout: lanes 0-15, VGPR r hold tq[t=r][d=dt*16+lane] (t<8 valid).
    if (lane < 16) {
      int dglob = dt * 16 + lane;
      float bias = bk[dglob];
      float vv[8];
#pragma unroll
      for (int v = 0; v < 8; ++v) vv[v] = vbase[(long long)v * DIM + dglob];
#pragma unroll
      for (int r = 0; r < 8; ++r) {
        float tq = acc[r] + bias;
#pragma unroll
        for (int v = 0; v < 8; ++v) atomicAdd(&lsim[r * 8 + v], tq * vv[v]);
      }
    }
  }
  __syncthreads();

  // Row softmax of the 8x8 sim matrix.
  if (tid < TOPK) {
    float mx = -INFINITY;
#pragma unroll
    for (int v = 0; v < 8; ++v) mx = fmaxf(mx, lsim[tid * 8 + v]);
    float sum = 0.f, e[8];
#pragma unroll
    for (int v = 0; v < 8; ++v) {
      e[v] = __expf(lsim[tid * 8 + v] - mx);
      sum += e[v];
    }
    float inv = 1.f / sum;
#pragma unroll
    for (int v = 0; v < 8; ++v) lhe[tid * 8 + v] = e[v] * inv;
  }
  __syncthreads();

  // text_out[t] = sum_v he[t][v]*visual[v]; visual_out[v] = sum_t he[t][v]*tk[t]
  for (int j = tid; j < TOPK * DIM; j += 512) {
    int t = j >> 10, d = j & (DIM - 1);
    float s0 = 0.f, s1 = 0.f;
#pragma unroll
    for (int v = 0; v < 8; ++v)
      s0 += lhe[t * 8 + v] * vbase[(long long)v * DIM + d];
    out_text[((long long)b * SEQ_LT + t) * DIM + d] = s0;
    int v = t;  // reuse j as (v,d) index for the visual output
#pragma unroll
    for (int t2 = 0; t2 < 8; ++t2) s1 += lhe[t2 * 8 + v] * ltk[t2 * DIM + d];
    out_vis[((long long)b * SEQ_LV + v) * DIM + d] = s1;
  }
}

extern "C" void kernel_launch(void* const* d_in, const int* in_sizes, int n_in,
                              void* d_out, int out_size, void* d_ws,
                              size_t ws_size, hipStream_t stream) {
  const float* text = (const float*)d_in[0];
  const float* visual = (const float*)d_in[1];
  const float* Wk = (const float*)d_in[2];
  const float* bk = (const float*)d_in[3];

  float* out = (float*)d_out;
  float* out_text = out;
  float* out_vis = out + (long long)BATCH * SEQ_LT * DIM;

  float* scores = (float*)d_ws;  // B*LT floats = 128 KB
  int* idx = (int*)((char*)d_ws + (size_t)BATCH * SEQ_LT * sizeof(float));

  long long n4 = (long long)out_size / 4;  // out_size floats -> v4f count
  zero_fill_kernel<<<4096, 256, 0, stream>>>((v4f*)d_out, n4);
  row_norm2_kernel<<<(BATCH * SEQ_LT) / 8, 256, 0, stream>>>(text, scores);
  topk_kernel<<<BATCH, 256, 0, stream>>>(scores, idx);
  fusion_kernel<<<BATCH, 512, 0, stream>>>(text, visual, Wk, bk, idx, out_text,
                                           out_vis);
}